// ScatterConnection_52767968199015
// MI455X (gfx1250) — compile-verified
//
#include <hip/hip_runtime.h>
#include <cstdint>

// ScatterConnection: out[b, n, y, x] += x[b, m, n], out zero-initialized.
// B=32, M=512, N=256, H=W=128. Pure bandwidth problem (~540MB traffic,
// ~23us floor at 23.3 TB/s). No matmul structure -> no WMMA; use NT b128
// stores for the 512MB zero-fill and gfx1250 async global->LDS staging.

typedef float v4f __attribute__((ext_vector_type(4)));

__global__ void ScatterConnection_zero_kernel(v4f* __restrict__ out, size_t n4) {
    size_t i      = (size_t)blockIdx.x * blockDim.x + threadIdx.x;
    size_t stride = (size_t)gridDim.x * blockDim.x;
    v4f z = {0.f, 0.f, 0.f, 0.f};
    for (; i < n4; i += stride) {
        __builtin_nontemporal_store(z, out + i);   // global_store_b128 ... nt
    }
}

// One block of 256 threads per entity (b, m).
__global__ void __launch_bounds__(256)
ScatterConnection_scatter_kernel(const float* __restrict__ x,
                                 const long long* __restrict__ loc,
                                 float* __restrict__ out,
                                 int N, int HW, int Wdim) {
    __shared__ float row[256];

    const int bm  = blockIdx.x;      // b*M + m
    const int tid = threadIdx.x;     // 0..N-1

    // ---- stage x[b, m, 0:256] (1 KB) into LDS via gfx1250 async copy ----
    // waves 0 and 1 (lanes 0..63) each move 16 bytes.
    const float* src = x + (size_t)bm * (size_t)N;   // uniform -> SGPR pair
    if (tid < 64) {
        unsigned lds_off = (unsigned)(size_t)&row[tid * 4]; // LDS byte offset
        unsigned voff    = (unsigned)(tid * 16);            // global byte offset
        asm volatile(
            "global_load_async_to_lds_b128 %0, %1, %2\n\t"
            "s_wait_asynccnt 0x0"
            :
            : "v"(lds_off), "v"(voff), "s"(src)
            : "memory");
    }
    __syncthreads();

    // ---- compute flat spatial index from (y, x) int64 pair ----
    const long long y  = loc[(size_t)bm * 2 + 0];
    const long long xx = loc[(size_t)bm * 2 + 1];
    const int flat = (int)(y * (long long)Wdim + xx);      // 0..HW-1

    // ---- scatter: one hardware fp32 atomic per (n) channel ----
    const int b = bm >> 9;                                  // bm / M, M=512
    const float v = row[tid];
    const size_t oidx = ((size_t)(b * N + tid) * (size_t)HW) + (size_t)flat;
    // relaxed agent-scope fetch-add, result unused -> no-return
    // global_atomic_add_f32 (STOREcnt-tracked)
    (void)__hip_atomic_fetch_add(&out[oidx], v,
                                 __ATOMIC_RELAXED, __HIP_MEMORY_SCOPE_AGENT);
}

extern "C" void kernel_launch(void* const* d_in, const int* in_sizes, int n_in,
                              void* d_out, int out_size, void* d_ws, size_t ws_size,
                              hipStream_t stream) {
    const float*     x   = (const float*)d_in[0];       // (B, M, N) f32
    const long long* loc = (const long long*)d_in[1];   // (B, M, 2) i64
    float*           out = (float*)d_out;               // (B, N, H, W) f32

    const int H = 128, W = 128, HW = H * W;
    const int BM = in_sizes[1] / 2;                     // B*M = 16384
    const int N  = in_sizes[0] / BM;                    // 256

    // 1) zero-fill 512MB output with NT b128 stores.
    size_t n4 = (size_t)out_size / 4;                   // float4 count
    ScatterConnection_zero_kernel<<<16384, 256, 0, stream>>>((v4f*)out, n4);

    // 2) scatter-add: one block per entity.
    ScatterConnection_scatter_kernel<<<BM, 256, 0, stream>>>(x, loc, out, N, HW, W);
}